// SeqAttnMatch_63694365000417
// MI455X (gfx1250) — compile-verified
//
#include <hip/hip_runtime.h>

// SeqAttnMatch, algebraically refactored for MI455X (gfx1250, wave32).
//   tp   = tweet @ Wk + bk                  (WMMA f32 GEMM, 2048x512x512)
//   q    = tp @ Wq^T                        (WMMA f32 GEMM; bq cancels in softmax)
//   s    = history . q  (masked rows skipped), softmax -> alpha
//   hbar = sum_l alpha * history[b,l]       (alpha==0 rows skipped; HBM-bound pass)
//   out  = relu((hbar@Wv+bv) @ W1 + b1) @ W2 + b2   (3 more WMMA f32 GEMMs)
// Needs 20 MB of workspace.

typedef __attribute__((ext_vector_type(2))) float v2f;
typedef __attribute__((ext_vector_type(8))) float v8f;

#define BDIM 2048
#define LDIM 200
#define EDIM 512
#define HDIM 512

// One wave per 16x16 C tile, K-loop of 4 using V_WMMA_F32_16X16X4_F32.
// B element address = Bm + k*bsK + n*bsN  (lets us feed Wq transposed).
__global__ __launch_bounds__(32) void gemm_wmma_f32(
    const float* __restrict__ A, const float* __restrict__ Bm,
    const float* __restrict__ bias, float* __restrict__ C,
    int M, int N, int K, int lda, long bsK, long bsN, int ldc, int do_relu)
{
    const int ntn = N >> 4;
    const int tm  = blockIdx.x / ntn;
    const int tn  = blockIdx.x - tm * ntn;
    const int m0  = tm << 4, n0 = tn << 4;
    const int lane = threadIdx.x & 31;
    const int hl   = lane >> 4;   // 0: K={0,1}, 1: K={2,3}
    const int l16  = lane & 15;

    const float* Arow = A  + (size_t)(m0 + l16) * lda;   // A: row = m0 + l16
    const float* Bcol = Bm + (size_t)(n0 + l16) * bsN;   // B: col = n0 + l16

    v8f acc = {0.f, 0.f, 0.f, 0.f, 0.f, 0.f, 0.f, 0.f};
#pragma unroll 4
    for (int k0 = 0; k0 < K; k0 += 4) {
        const int ka = k0 + (hl << 1);
        v2f a, b;
        a.x = Arow[ka];
        a.y = Arow[ka + 1];
        const float* Bp = Bcol + (size_t)ka * bsK;
        b.x = Bp[0];
        b.y = Bp[bsK];
        acc = __builtin_amdgcn_wmma_f32_16x16x4_f32(
            false, a, false, b, (short)0, acc, false, false);
    }

    const int col = n0 + l16;
    const float bb = bias ? bias[col] : 0.f;
#pragma unroll
    for (int r = 0; r < 8; ++r) {
        float v = acc[r] + bb;
        if (do_relu) v = fmaxf(v, 0.f);
        const int row = m0 + r + (hl << 3);   // C: VGPR r -> M = r + 8*(lane>=16)
        C[(size_t)row * ldc + col] = v;
    }
}

// One 8-wave block per batch. Pass 1: wave-per-row dot(history[b,l], q[b])
// skipping masked rows (mask cuts ~50% of HBM traffic). Block softmax in LDS.
// Pass 2: coalesced alpha-weighted sum over e, skipping alpha==0 rows.
__global__ __launch_bounds__(256) void attn_kernel(
    const float* __restrict__ history, const unsigned char* __restrict__ mask,
    const float* __restrict__ q, float* __restrict__ hbar)
{
    __shared__ float qs[EDIM];
    __shared__ float sc[256];
    __shared__ float red[8];

    const int b    = blockIdx.x;
    const int tid  = threadIdx.x;
    const int wv   = tid >> 5;
    const int lane = tid & 31;

    qs[tid]       = q[(size_t)b * EDIM + tid];
    qs[tid + 256] = q[(size_t)b * EDIM + tid + 256];
    if (tid >= LDIM) sc[tid] = -3.0e38f;   // pad so block reductions run over 256
    __syncthreads();

    const float* hb = history + (size_t)b * LDIM * EDIM;
    const unsigned char* mb = mask + (size_t)b * LDIM;

    for (int l = wv; l < LDIM; l += 8) {
        if (mb[l]) {                       // padded position: score is -inf anyway
            if (lane == 0) sc[l] = -3.0e38f;
            continue;
        }
        const float* hrow = hb + (size_t)l * EDIM;
        __builtin_prefetch(hrow + 8 * EDIM, 0, 1);   // next row for this wave
        float s = 0.f;
#pragma unroll
        for (int i = 0; i < 16; ++i) {
            const int e = lane + (i << 5);
            s = fmaf(hrow[e], qs[e], s);
        }
#pragma unroll
        for (int off = 16; off > 0; off >>= 1)
            s += __shfl_xor(s, off, 32);
        if (lane == 0) sc[l] = s;
    }
    __syncthreads();

    // Softmax over sc[0..255] (padding at -3e38 contributes exp()=0).
    const float v = sc[tid];
    float m = v;
#pragma unroll
    for (int off = 16; off > 0; off >>= 1)
        m = fmaxf(m, __shfl_xor(m, off, 32));
    if (lane == 0) red[wv] = m;
    __syncthreads();
    float mx = red[0];
#pragma unroll
    for (int i = 1; i < 8; ++i) mx = fmaxf(mx, red[i]);

    const float ex = __expf(v - mx);
    float s = ex;
#pragma unroll
    for (int off = 16; off > 0; off >>= 1)
        s += __shfl_xor(s, off, 32);
    __syncthreads();
    if (lane == 0) red[wv] = s;
    __syncthreads();
    float tot = 0.f;
#pragma unroll
    for (int i = 0; i < 8; ++i) tot += red[i];
    sc[tid] = ex / tot;                    // alpha (0 for masked / padded)
    __syncthreads();

    float acc0 = 0.f, acc1 = 0.f;
    for (int l = 0; l < LDIM; ++l) {
        const float a = sc[l];
        if (a == 0.f) continue;            // uniform branch: skip masked rows
        const float* hrow = hb + (size_t)l * EDIM;
        acc0 = fmaf(a, hrow[tid], acc0);
        acc1 = fmaf(a, hrow[tid + 256], acc1);
    }
    hbar[(size_t)b * EDIM + tid]       = acc0;
    hbar[(size_t)b * EDIM + tid + 256] = acc1;
}

extern "C" void kernel_launch(void* const* d_in, const int* in_sizes, int n_in,
                              void* d_out, int out_size, void* d_ws, size_t ws_size,
                              hipStream_t stream)
{
    (void)in_sizes; (void)n_in; (void)out_size; (void)ws_size;
    const float* tweet   = (const float*)d_in[0];
    const float* history = (const float*)d_in[1];
    const unsigned char* hmask = (const unsigned char*)d_in[2];
    const float* Wq = (const float*)d_in[3];
    // d_in[4] = bq: constant over l -> cancels inside softmax, unused.
    const float* Wk = (const float*)d_in[5];
    const float* bk = (const float*)d_in[6];
    const float* Wv = (const float*)d_in[7];
    const float* bv = (const float*)d_in[8];
    const float* W1 = (const float*)d_in[9];
    const float* b1 = (const float*)d_in[10];
    const float* W2 = (const float*)d_in[11];
    const float* b2 = (const float*)d_in[12];
    float* out = (float*)d_out;

    char* ws = (char*)d_ws;
    const size_t MAT = (size_t)BDIM * HDIM * sizeof(float);   // 4 MB each
    float* tp      = (float*)(ws + 0 * MAT);
    float* qv      = (float*)(ws + 1 * MAT);
    float* hbar    = (float*)(ws + 2 * MAT);
    float* matched = (float*)(ws + 3 * MAT);
    float* h1      = (float*)(ws + 4 * MAT);

    const dim3 gg((BDIM / 16) * (HDIM / 16));   // 4096 waves, one 16x16 tile each
    const dim3 bw(32);

    // 1. tp = tweet @ Wk + bk            (Wk[e,h]: bsK=H, bsN=1)
    hipLaunchKernelGGL(gemm_wmma_f32, gg, bw, 0, stream,
        tweet, Wk, bk, tp, BDIM, HDIM, EDIM, EDIM, (long)HDIM, 1L, HDIM, 0);
    // 2. qv = tp @ Wq^T                  (B[k,n] = Wq[n,k]: bsK=1, bsN=H)
    hipLaunchKernelGGL(gemm_wmma_f32, gg, bw, 0, stream,
        tp, Wq, (const float*)nullptr, qv, BDIM, EDIM, HDIM, HDIM, 1L, (long)HDIM, EDIM, 0);
    // 3. masked scores -> softmax -> hbar (the only HBM-heavy pass)
    hipLaunchKernelGGL(attn_kernel, dim3(BDIM), dim3(256), 0, stream,
        history, hmask, qv, hbar);
    // 4. matched = hbar @ Wv + bv
    hipLaunchKernelGGL(gemm_wmma_f32, gg, bw, 0, stream,
        hbar, Wv, bv, matched, BDIM, HDIM, EDIM, EDIM, (long)HDIM, 1L, HDIM, 0);
    // 5. h1 = relu(matched @ W1 + b1)
    hipLaunchKernelGGL(gemm_wmma_f32, gg, bw, 0, stream,
        matched, W1, b1, h1, BDIM, HDIM, HDIM, HDIM, (long)HDIM, 1L, HDIM, 1);
    // 6. out = h1 @ W2 + b2
    hipLaunchKernelGGL(gemm_wmma_f32, gg, bw, 0, stream,
        h1, W2, b2, out, BDIM, HDIM, HDIM, HDIM, (long)HDIM, 1L, HDIM, 0);
}